// Model_23880018165862
// MI455X (gfx1250) — compile-verified
//
#include <hip/hip_runtime.h>
#include <math.h>

// CDNA5 (gfx1250) wave32 WMMA types
typedef float v2f __attribute__((ext_vector_type(2)));
typedef float v8f __attribute__((ext_vector_type(8)));

#define FEAT 64
#define WAVES_PER_BLOCK 8

// One wave (32 lanes) per destination node.
// Segment softmax over the node's edges, then weighted aggregation of gathered
// source rows done with V_WMMA_F32_16X16X4_F32 (full f32 precision):
//   out(1x64) = w(1x16) . V(16x64)
// tiled as 4 N-chunks (16 features) x 4 K-chunks (4 neighbors), C-accumulated.
__global__ __launch_bounds__(32 * WAVES_PER_BLOCK)
void gat_softmax_aggregate(const int* __restrict__ row_ptr,
                           const int* __restrict__ col_idx,
                           const float* __restrict__ scores,
                           const float* __restrict__ value,
                           float* __restrict__ out,
                           int num_nodes)
{
    const int lane = threadIdx.x & 31;
    const int wave = threadIdx.x >> 5;
    const int node = blockIdx.x * WAVES_PER_BLOCK + wave;
    if (node >= num_nodes) return;   // wave-uniform: EXEC stays all-ones below

    const int l16  = lane & 15;      // logical edge slot (duplicated per half)
    const int half = lane >> 4;      // 0: lanes 0-15, 1: lanes 16-31

    const int start = row_ptr[node];
    const int end   = row_ptr[node + 1];

    // ---- pass 1: segment max (both halves hold duplicate data; xor<16 reduce)
    float m = -INFINITY;
    for (int e = start + l16; e < end; e += 16) m = fmaxf(m, scores[e]);
#pragma unroll
    for (int d = 1; d < 16; d <<= 1) m = fmaxf(m, __shfl_xor(m, d, 32));

    // ---- pass 2: sum of exp(s - max)
    float ssum = 0.f;
    for (int e = start + l16; e < end; e += 16) ssum += expf(scores[e] - m);
#pragma unroll
    for (int d = 1; d < 16; d <<= 1) ssum += __shfl_xor(ssum, d, 32);
    const float inv = (ssum > 0.f) ? (1.f / ssum) : 0.f;

    // ---- pass 3: weighted gather-aggregate via WMMA f32 16x16x4
    v8f acc[4];
#pragma unroll
    for (int g = 0; g < 4; ++g) acc[g] = (v8f){0.f,0.f,0.f,0.f,0.f,0.f,0.f,0.f};

    for (int e0 = start; e0 < end; e0 += 16) {
        const int  e     = e0 + l16;
        const bool valid = (e < end);
        // softmax weight for my edge slot (0 for padding slots)
        const float w = valid ? expf(scores[e] - m) * inv : 0.f;
        const int   c = valid ? col_idx[e] : 0;   // padded rows multiplied by w=0

#pragma unroll
        for (int kc = 0; kc < 4; ++kc) {
            // ISA layout for 32-bit 16x4 A / 4x16 B: K = vgpr_idx + 2*half
            const int k0 = kc * 4 + half * 2;
            v2f a;                               // A rows all identical = weights
            a.x = __shfl(w, k0,     32);
            a.y = __shfl(w, k0 + 1, 32);
            const long r0 = (long)__shfl(c, k0,     32) * FEAT;
            const long r1 = (long)__shfl(c, k0 + 1, 32) * FEAT;

#pragma unroll
            for (int g = 0; g < 4; ++g) {        // 16-feature N-chunks
                v2f b;                            // B[k][n]: n = l16 within chunk
                b.x = value[r0 + g * 16 + l16];
                b.y = value[r1 + g * 16 + l16];
                // 8 args: (neg_a, A, neg_b, B, c_mod, C, reuse_a, reuse_b)
                acc[g] = __builtin_amdgcn_wmma_f32_16x16x4_f32(
                    false, a, false, b, (short)0, acc[g], false, false);
            }
        }
    }

    // ---- store: all D rows identical; lanes 0-15 see row 0, lanes 16-31 row 8.
    // Two fully-coalesced 128B stores covering features 0..31 and 32..63.
    float* o = out + (long)node * FEAT;
    o[lane]      = half ? acc[1][0] : acc[0][0];   // f = 0..31
    o[lane + 32] = half ? acc[3][0] : acc[2][0];   // f = 32..63
}

extern "C" void kernel_launch(void* const* d_in, const int* in_sizes, int n_in,
                              void* d_out, int out_size, void* d_ws, size_t ws_size,
                              hipStream_t stream)
{
    const int*   row_ptr = (const int*)d_in[0];    // [N+1]
    const int*   col_idx = (const int*)d_in[1];    // [E]
    const float* scores  = (const float*)d_in[2];  // [E]
    const float* value   = (const float*)d_in[3];  // [N, 64]
    float*       out     = (float*)d_out;          // [N, 64]

    const int num_nodes = in_sizes[0] - 1;
    const dim3 block(32 * WAVES_PER_BLOCK);
    const dim3 grid((num_nodes + WAVES_PER_BLOCK - 1) / WAVES_PER_BLOCK);
    gat_softmax_aggregate<<<grid, block, 0, stream>>>(row_ptr, col_idx, scores,
                                                      value, out, num_nodes);
}